// VL20_22969485099771
// MI455X (gfx1250) — compile-verified
//
#include <hip/hip_runtime.h>
#include <hip/hip_bf16.h>
#include <stdint.h>

// Two-layer Lorenz-96 RHS. Memory-bound stream: ~302MB @ 23.3TB/s -> ~13us floor.
// Strategy: coalesced async global->LDS tile staging (CDNA5 async data mover),
// per-thread stencil compute from padded LDS, coalesced async LDS->global out.

#define NX 36
#define NF 72          // floats per batch (2 layers x 36)
#define TILE 64        // batches per workgroup == threads per workgroup
#define PADF 76        // padded floats per batch in LDS (304B: 16B-aligned, good bank spread)
#define F4B 18         // float4s per batch (72/4)
#define ITERS 18       // (TILE*F4B)/TILE coalesced float4 iterations

#if defined(__gfx1250__) &&                                                    \
    __has_builtin(__builtin_amdgcn_global_load_async_to_lds_b128) &&           \
    __has_builtin(__builtin_amdgcn_global_store_async_from_lds_b128) &&        \
    __has_builtin(__builtin_amdgcn_s_wait_asynccnt)
#define USE_ASYNC 1
#else
#define USE_ASYNC 0
#endif

// Builtin parameter types (from hipcc diagnostic): param1 is a 16B int-vector
// pointer in the global (__device__/AS1) address space.
typedef int v4i __attribute__((vector_size(4 * sizeof(int))));
typedef __attribute__((address_space(1))) v4i GV4;  // global float4-as-int4
typedef __attribute__((address_space(3))) v4i LV4;  // LDS float4-as-int4

__global__ __launch_bounds__(TILE) void l96_tile_kernel(const float* __restrict__ in,
                                                        float* __restrict__ out) {
  __shared__ __align__(16) float sIn[TILE * PADF];
  __shared__ __align__(16) float sOut[TILE * PADF];

  const int tid = threadIdx.x;
  const long tileF = (long)blockIdx.x * (TILE * NF);  // float offset of this tile

  // ---- Phase 1: coalesced global -> LDS (async data mover on gfx1250) ----
#pragma unroll
  for (int k = 0; k < ITERS; ++k) {
    const int f4 = tid + k * TILE;         // float4 index within tile (coalesced)
    const int b = f4 / F4B;                // batch within tile
    const int e = f4 - b * F4B;            // float4 within batch
    const int ldsOff = b * PADF + e * 4;   // float index (16B aligned)
#if USE_ASYNC
    __builtin_amdgcn_global_load_async_to_lds_b128(
        (GV4*)(in + tileF + (long)f4 * 4), (LV4*)(sIn + ldsOff), 0, 0);
#else
    const float4 v = *reinterpret_cast<const float4*>(in + tileF + (long)f4 * 4);
    *reinterpret_cast<float4*>(sIn + ldsOff) = v;
#endif
  }
#if USE_ASYNC
  __builtin_amdgcn_s_wait_asynccnt(0);
#endif
  __syncthreads();

  // ---- Phase 2: per-thread stencil over its batch (all regs, full unroll) ----
  {
    const int base = tid * PADF;
    float X[NX], T[NX];
#pragma unroll
    for (int j = 0; j < NX; ++j) X[j] = sIn[base + j];
#pragma unroll
    for (int j = 0; j < NX; ++j) T[j] = sIn[base + NX + j];

#pragma unroll
    for (int j = 0; j < NX; ++j) {
      const int jm1 = (j + NX - 1) % NX;
      const int jm2 = (j + NX - 2) % NX;
      const int jp1 = (j + 1) % NX;
      const int jp2 = (j + 2) % NX;
      const float xm1 = X[jm1], xp1 = X[jp1];
      // dX = (X[j-1]-X[j+2])*X[j+1] - GAMMA*X[j] + F - ALPHA*T[j]
      sOut[base + j] = (xm1 - X[jp2]) * xp1 - X[j] + 10.0f - T[j];
      // dT = X[j-1]*T[j-2] - X[j+1]*T[j+2] - GAMMA*T[j] + ALPHA*X[j] + G
      sOut[base + NX + j] = xm1 * T[jm2] - xp1 * T[jp2] - T[j] + X[j] + 10.0f;
    }
  }
  __syncthreads();

  // ---- Phase 3: coalesced LDS -> global (async store-from-LDS) ----
#pragma unroll
  for (int k = 0; k < ITERS; ++k) {
    const int f4 = tid + k * TILE;
    const int b = f4 / F4B;
    const int e = f4 - b * F4B;
    const int ldsOff = b * PADF + e * 4;
#if USE_ASYNC
    __builtin_amdgcn_global_store_async_from_lds_b128(
        (GV4*)(out + tileF + (long)f4 * 4), (LV4*)(sOut + ldsOff), 0, 0);
#else
    const float4 v = *reinterpret_cast<const float4*>(sOut + ldsOff);
    *reinterpret_cast<float4*>(out + tileF + (long)f4 * 4) = v;
#endif
  }
#if USE_ASYNC
  __builtin_amdgcn_s_wait_asynccnt(0);  // S_ENDPGM also waits, this is belt+braces
#endif
}

// Tail kernel for batch counts not divisible by TILE (not hit for B=524288).
__global__ void l96_tail_kernel(const float* __restrict__ in, float* __restrict__ out,
                                int startBatch, int nBatch) {
  const int b = startBatch + blockIdx.x * blockDim.x + threadIdx.x;
  if (b >= nBatch) return;
  const float* X = in + (long)b * NF;
  const float* T = X + NX;
  float* dX = out + (long)b * NF;
  float* dT = dX + NX;
#pragma unroll
  for (int j = 0; j < NX; ++j) {
    const int jm1 = (j + NX - 1) % NX;
    const int jm2 = (j + NX - 2) % NX;
    const int jp1 = (j + 1) % NX;
    const int jp2 = (j + 2) % NX;
    const float xm1 = X[jm1], xp1 = X[jp1];
    dX[j] = (xm1 - X[jp2]) * xp1 - X[j] + 10.0f - T[j];
    dT[j] = xm1 * T[jm2] - xp1 * T[jp2] - T[j] + X[j] + 10.0f;
  }
}

extern "C" void kernel_launch(void* const* d_in, const int* in_sizes, int n_in,
                              void* d_out, int out_size, void* d_ws, size_t ws_size,
                              hipStream_t stream) {
  const float* x = (const float*)d_in[0];  // (B, 2, NX) float32
  float* out = (float*)d_out;              // (B, 2, NX) float32

  const int nBatch = in_sizes[0] / NF;     // 524288
  const int nTiles = nBatch / TILE;        // 8192
  const int rem = nBatch - nTiles * TILE;  // 0 for the reference shape

  if (nTiles > 0) {
    l96_tile_kernel<<<nTiles, TILE, 0, stream>>>(x, out);
  }
  if (rem > 0) {
    const int start = nTiles * TILE;
    l96_tail_kernel<<<(rem + 63) / 64, 64, 0, stream>>>(x, out, start, nBatch);
  }
}